// GraphSAGE_69020124447114
// MI455X (gfx1250) — compile-verified
//
#include <hip/hip_runtime.h>
#include <hip/hip_bf16.h>

typedef __attribute__((ext_vector_type(2))) float v2f;
typedef __attribute__((ext_vector_type(8))) float v8f;

// ---------------------------------------------------------------------------
// Degree: one thread per edge, f32 atomic histogram of dst.
// ---------------------------------------------------------------------------
__global__ void degree_kernel(const long long* __restrict__ dst,
                              float* __restrict__ deg, long long n_edges) {
    long long e = (long long)blockIdx.x * blockDim.x + threadIdx.x;
    if (e < n_edges) {
        atomicAdd(&deg[dst[e]], 1.0f);
    }
}

// ---------------------------------------------------------------------------
// Scatter-sum: C/4 lanes per edge; float4 gather from feat[src], four
// global_atomic_add_f32 into agg[dst]. Feature tables are L2-resident
// (x: 25.6MB, h: 51.2MB vs 192MB L2) -> atomics resolve at L2.
// ---------------------------------------------------------------------------
template <int C>
__global__ void scatter_sum_kernel(const float* __restrict__ feat,
                                   const long long* __restrict__ src,
                                   const long long* __restrict__ dst,
                                   float* __restrict__ agg, long long n_edges) {
    constexpr int TPE = C / 4;  // threads per edge
    long long gid = (long long)blockIdx.x * blockDim.x + threadIdx.x;
    long long e = gid / TPE;
    if (e >= n_edges) return;
    int c4 = (int)(gid % TPE) * 4;
    long long s = src[e];
    long long d = dst[e];
    float4 v = *(const float4*)(feat + (size_t)s * C + c4);
    float* p = agg + (size_t)d * C + c4;
    atomicAdd(p + 0, v.x);
    atomicAdd(p + 1, v.y);
    atomicAdd(p + 2, v.z);
    atomicAdd(p + 3, v.w);
}

// ---------------------------------------------------------------------------
// Fused SAGE layer (or plain projection when DUAL=false):
//   DUAL:  out = relu?( (agg/max(deg,1)) @ Wl + bl + feat @ Wr )
//   !DUAL: out = feat @ Wl + bl
// One block per 16-node tile. NOUT/16 waves per block; each wave owns a
// 16-column output slice and runs V_WMMA_F32_16X16X4_F32 over K in steps
// of 4. A fragments come from LDS-staged tiles, B fragments from the
// (L2-hot) weight matrices. Full tiles (all but the last block) take a
// completely branch-free load + store path.
// ---------------------------------------------------------------------------
template <int K, int NOUT, bool DUAL, bool RELU>
__global__ __launch_bounds__(2 * NOUT) void sage_layer_kernel(
    const float* __restrict__ feat,  // [n, K]
    const float* __restrict__ agg,   // [n, K] summed neighbor feats (DUAL)
    const float* __restrict__ deg,   // [n]               (DUAL)
    const float* __restrict__ Wl,    // [K, NOUT]
    const float* __restrict__ bl,    // [NOUT]
    const float* __restrict__ Wr,    // [K, NOUT]         (DUAL)
    float* __restrict__ out,         // [n, NOUT]
    int n) {
    constexpr int KP = K + 4;  // padded LDS row stride: float4-aligned, no bank conflicts
    __shared__ float sMean[16 * KP];
    __shared__ float sFeat[DUAL ? 16 * KP : 4];

    const int tid = threadIdx.x;
    const int base = blockIdx.x * 16;
    const bool full = (base + 16 <= n);  // uniform per block

    // ---- cooperative stage of the 16xK tile(s) into LDS (coalesced float4)
    constexpr int TOTAL4 = 16 * K / 4;
    for (int idx = tid; idx < TOTAL4; idx += blockDim.x) {
        const int row = idx / (K / 4);
        const int c4 = (idx % (K / 4)) * 4;
        const int node = base + row;
        float4 a = make_float4(0.f, 0.f, 0.f, 0.f);
        float4 b = make_float4(0.f, 0.f, 0.f, 0.f);
        if (full || node < n) {
            if constexpr (DUAL) {
                float4 s = *(const float4*)(agg + (size_t)node * K + c4);
                float inv = 1.0f / fmaxf(deg[node], 1.0f);
                a = make_float4(s.x * inv, s.y * inv, s.z * inv, s.w * inv);
                b = *(const float4*)(feat + (size_t)node * K + c4);
            } else {
                a = *(const float4*)(feat + (size_t)node * K + c4);
            }
        }
        *(float4*)&sMean[row * KP + c4] = a;
        if constexpr (DUAL) *(float4*)&sFeat[row * KP + c4] = b;
    }
    __syncthreads();

    const int lane = tid & 31;
    const int wave = tid >> 5;
    const int half = lane >> 4;  // 0: K={k,k+1}, 1: K={k+2,k+3}
    const int m = lane & 15;
    const int col = wave * 16 + m;

    v8f acc = {};
#pragma unroll 4
    for (int k0 = 0; k0 < K; k0 += 4) {
        const int ka = k0 + 2 * half;
        // A fragment (16x4 f32 layout): lane m holds row m, cols ka..ka+1
        v2f a;
        a.x = sMean[m * KP + ka];
        a.y = sMean[m * KP + ka + 1];
        // B fragment (4x16): lane holds B[ka..ka+1][col]
        v2f b;
        b.x = Wl[(size_t)ka * NOUT + col];
        b.y = Wl[(size_t)(ka + 1) * NOUT + col];
        acc = __builtin_amdgcn_wmma_f32_16x16x4_f32(false, a, false, b,
                                                    (short)0, acc, false, false);
        if constexpr (DUAL) {
            v2f a2;
            a2.x = sFeat[m * KP + ka];
            a2.y = sFeat[m * KP + ka + 1];
            v2f b2;
            b2.x = Wr[(size_t)ka * NOUT + col];
            b2.y = Wr[(size_t)(ka + 1) * NOUT + col];
            acc = __builtin_amdgcn_wmma_f32_16x16x4_f32(
                false, a2, false, b2, (short)0, acc, false, false);
        }
    }

    // ---- epilogue: bias + optional relu, store per 16x16 f32 C/D layout:
    // VGPR r, lanes 0-15 -> M=r, lanes 16-31 -> M=r+8.
    const float bias = bl[col];
    float res[8];
#pragma unroll
    for (int r = 0; r < 8; ++r) {
        float v = acc[r] + bias;
        res[r] = RELU ? fmaxf(v, 0.0f) : v;
    }

    if (full) {
        // Branch-free fast path: one base pointer, immediate row offsets.
        float* p = out + (size_t)(base + 8 * half) * NOUT + col;
#pragma unroll
        for (int r = 0; r < 8; ++r) {
            p[(size_t)r * NOUT] = res[r];
        }
    } else {
#pragma unroll
        for (int r = 0; r < 8; ++r) {
            const int node = base + r + 8 * half;
            if (node < n) out[(size_t)node * NOUT + col] = res[r];
        }
    }
}

// ---------------------------------------------------------------------------
// Host-side orchestration (all on `stream`; graph-capture safe).
// ---------------------------------------------------------------------------
extern "C" void kernel_launch(void* const* d_in, const int* in_sizes, int n_in,
                              void* d_out, int out_size, void* d_ws,
                              size_t ws_size, hipStream_t stream) {
    constexpr int IN_C = 64, HID_C = 128, OUT_C = 64;

    const float* x = (const float*)d_in[0];
    const long long* ei = (const long long*)d_in[1];  // [2, E] int64
    const float* Wl0 = (const float*)d_in[2];
    const float* Wr0 = (const float*)d_in[3];
    const float* bl0 = (const float*)d_in[4];
    const float* Wl1 = (const float*)d_in[5];
    const float* Wr1 = (const float*)d_in[6];
    const float* bl1 = (const float*)d_in[7];
    const float* Wout = (const float*)d_in[8];
    const float* bout = (const float*)d_in[9];

    const int n = in_sizes[0] / IN_C;
    const long long E = (long long)(in_sizes[1] / 2);
    const long long* src = ei;
    const long long* dst = ei + E;

    // Workspace layout (floats): deg[n_pad] | agg[n_pad*128] | h[n_pad*128]
    size_t n_pad = ((size_t)n + 63) & ~(size_t)63;
    float* deg = (float*)d_ws;
    float* agg = deg + n_pad;
    float* h = agg + n_pad * (size_t)HID_C;

    const int nTiles = (n + 15) / 16;

    // ---- degree (shared by both layers) + layer0 aggregation buffer ----
    hipMemsetAsync(deg, 0, n_pad * sizeof(float), stream);
    hipMemsetAsync(agg, 0, (size_t)n * IN_C * sizeof(float), stream);

    {
        int blocks = (int)((E + 255) / 256);
        degree_kernel<<<blocks, 256, 0, stream>>>(dst, deg, E);
    }
    {
        long long threads = E * (IN_C / 4);
        int blocks = (int)((threads + 255) / 256);
        scatter_sum_kernel<IN_C><<<blocks, 256, 0, stream>>>(x, src, dst, agg, E);
    }
    // ---- layer 0: h = relu(mean @ Wl0 + bl0 + x @ Wr0) ----
    sage_layer_kernel<IN_C, HID_C, true, true>
        <<<nTiles, 2 * HID_C, 0, stream>>>(x, agg, deg, Wl0, bl0, Wr0, h, n);

    // ---- layer 1 aggregation ----
    hipMemsetAsync(agg, 0, (size_t)n * HID_C * sizeof(float), stream);
    {
        long long threads = E * (HID_C / 4);
        int blocks = (int)((threads + 255) / 256);
        scatter_sum_kernel<HID_C><<<blocks, 256, 0, stream>>>(h, src, dst, agg, E);
    }
    // ---- layer 1: h2 = relu(mean @ Wl1 + bl1 + h @ Wr1) ----
    // Output aliases `agg`: each block stages its 16 rows into LDS before any
    // store, and only ever writes its own rows -> no cross-block hazard.
    sage_layer_kernel<HID_C, HID_C, true, true>
        <<<nTiles, 2 * HID_C, 0, stream>>>(h, agg, deg, Wl1, bl1, Wr1, agg, n);

    // ---- output projection: out = h2 @ Wout + bout ----
    sage_layer_kernel<HID_C, OUT_C, false, false>
        <<<nTiles, 2 * OUT_C, 0, stream>>>(agg, nullptr, nullptr, Wout, bout,
                                           nullptr, (float*)d_out, n);
}